// GCNClassifier_41497974014275
// MI455X (gfx1250) — compile-verified
//
#include <hip/hip_runtime.h>
#include <math.h>
#include <stddef.h>

// ---------------------------------------------------------------------------
// GCN forward for MI455X (gfx1250, wave32).
//  - edge scatter/scatter-add: memory/atomic bound (~1.6 GB through L2)
//  - dense 128x128 GEMMs: V_WMMA_F32_16X16X4_F32, one 16x16 tile per wave,
//    operands staged into LDS with GLOBAL_LOAD_ASYNC_TO_LDS_B32 (ASYNCcnt)
// ---------------------------------------------------------------------------

typedef __attribute__((ext_vector_type(2))) float v2f;
typedef __attribute__((ext_vector_type(8))) float v8f;

#define HIDF 128
#define WT_STRIDE 130   // even (aligned ds_load_b64) + bank-conflict padding
#define A_STRIDE  132   // even + bank-conflict padding

// ---------------- utility kernels ----------------

__global__ void fill_zero_kernel(float* __restrict__ p, int n) {
    int i = blockIdx.x * blockDim.x + threadIdx.x;
    if (i < n) p[i] = 0.0f;
}

__global__ void degree_kernel(const int* __restrict__ src, const int* __restrict__ dst,
                              float* __restrict__ deg_out, float* __restrict__ deg_in, int nE) {
    int e = blockIdx.x * blockDim.x + threadIdx.x;
    if (e < nE) {
        atomicAdd(&deg_out[src[e]], 1.0f);
        atomicAdd(&deg_in[dst[e]], 1.0f);
    }
}

// in-place: d = clip(d, 1)^-0.5   (applied to both degree arrays at once)
__global__ void rsqrt_clip_kernel(float* __restrict__ p, int n) {
    int i = blockIdx.x * blockDim.x + threadIdx.x;
    if (i < n) p[i] = rsqrtf(fmaxf(p[i], 1.0f));
}

// ---------------- edge scatter-add ----------------
// 32 lanes per edge, float4 (16B) per lane -> 512B coalesced load per wave,
// 4 global_atomic_add_f32 per lane into the L2-resident accumulator.
// SCALE_SRC fuses the source-side D_out^{-1/2} normalization into the gather.
template <bool SCALE_SRC>
__global__ void __launch_bounds__(256)
scatter_add_kernel(const int* __restrict__ src, const int* __restrict__ dst,
                   const float* __restrict__ x, const float* __restrict__ srcScale,
                   float* __restrict__ agg, int nE) {
    int e = blockIdx.x * 8 + (threadIdx.x >> 5);
    if (e >= nE) return;
    int c = (threadIdx.x & 31) * 4;
    int s = src[e];
    int d = dst[e];
    float4 v = *(const float4*)&x[(size_t)s * HIDF + c];
    if (SCALE_SRC) {
        const float sc = srcScale[s];   // wave-uniform, L2-hot
        v.x *= sc; v.y *= sc; v.z *= sc; v.w *= sc;
    }
    float* p = &agg[(size_t)d * HIDF + c];
    atomicAdd(p + 0, v.x);
    atomicAdd(p + 1, v.y);
    atomicAdd(p + 2, v.z);
    atomicAdd(p + 3, v.w);
}

// ---------------- WMMA GEMM ----------------
// Out[row, col] = epilogue( rowScaleIn[row] * (A @ W)[row, col] + bias[col] )
// (diagonal row scaling commutes through the matmul, so it is fused into the
//  epilogue FMA and both LDS stagings are pure async copies)
// epilogue = ReLU, then (SCALE_OUT) * rowScaleOut[row].
// Block: 256 threads = 8 waves; block computes 16 rows x 128 cols; wave w owns
// column tile [16w,16w+16). K=128 in steps of 4 via v_wmma_f32_16x16x4_f32.
template <bool SCALE_OUT>
__global__ void __launch_bounds__(256)
gemm16_wmma_kernel(const float* __restrict__ A, const float* __restrict__ rowScaleIn,
                   const float* __restrict__ W, const float* __restrict__ bias,
                   const float* __restrict__ rowScaleOut, float* __restrict__ Out,
                   int nrows) {
    __shared__ float sWt[HIDF * WT_STRIDE];  // W transposed: sWt[col*WT_STRIDE + k]
    __shared__ float sA[16 * A_STRIDE];      // raw A tile:   sA[m*A_STRIDE + k]

    const int tid  = threadIdx.x;
    const int row0 = blockIdx.x * 16;

    // Workgroup-relative LDS byte offsets (generic LDS address carries the
    // LDS offset in addr[31:0] per the gfx1250 aperture encoding).
    const unsigned wtBase = (unsigned)(size_t)&sWt[0];
    const unsigned aBase  = (unsigned)(size_t)&sA[0];

    // Async transpose-stage of W: coalesced b32 global reads per lane, per-lane
    // strided LDS destinations. Tracked by ASYNCcnt, VALU stays free.
    for (int i = tid; i < HIDF * HIDF; i += 256) {
        int k = i >> 7;
        int c = i & 127;
        unsigned ldsoff = wtBase + 4u * (unsigned)(c * WT_STRIDE + k);
        const float* g = W + i;
        asm volatile("global_load_async_to_lds_b32 %0, %1, off"
                     :: "v"(ldsoff), "v"(g) : "memory");
    }
    // Async stage of the 16x128 A tile (pure copy; padding via per-lane dests).
    for (int i = tid; i < 16 * HIDF; i += 256) {
        int m = i >> 7;
        int c = i & 127;
        int row = row0 + m;
        row = row < nrows ? row : nrows - 1;   // clamp (N % 16 == 0 anyway)
        unsigned ldsoff = aBase + 4u * (unsigned)(m * A_STRIDE + c);
        const float* g = A + (size_t)row * HIDF + c;
        asm volatile("global_load_async_to_lds_b32 %0, %1, off"
                     :: "v"(ldsoff), "v"(g) : "memory");
    }
    asm volatile("s_wait_asynccnt 0x0" ::: "memory");  // my async writes landed
    __syncthreads();                                   // everyone's landed

    const int wv   = tid >> 5;       // wave id -> column tile
    const int lane = tid & 31;
    const int half = lane >> 4;      // 0: K pair {k,k+1}, 1: {k+2,k+3}
    const int lm   = lane & 15;
    const int col  = wv * 16 + lm;

    // 32-bit A 16x4 layout: lanes 0-15 hold A[lm][k..k+1], lanes 16-31 A[lm][k+2..k+3]
    // 32-bit B 4x16 layout: lanes 0-15 hold B[k..k+1][col], lanes 16-31 B[k+2..k+3][col]
    v8f acc = {};
#pragma unroll 4
    for (int k = 0; k < HIDF; k += 4) {
        const int kb = k + 2 * half;
        v2f a = *(const v2f*)&sA[lm * A_STRIDE + kb];
        v2f b = *(const v2f*)&sWt[col * WT_STRIDE + kb];
        acc = __builtin_amdgcn_wmma_f32_16x16x4_f32(
            /*neg_a=*/false, a, /*neg_b=*/false, b,
            /*c_mod=*/(short)0, acc, /*reuse_a=*/false, /*reuse_b=*/false);
    }

    // C/D layout: VGPR v -> M = v (lanes 0-15) / v+8 (lanes 16-31), N = lane&15.
    const float bcol = bias[col];
    if (row0 + 16 <= nrows) {          // uniform full-tile fast path
#pragma unroll
        for (int v = 0; v < 8; ++v) {
            const int row = row0 + v + 8 * half;
            float val = fmaxf(fmaf(acc[v], rowScaleIn[row], bcol), 0.0f);
            if (SCALE_OUT) val *= rowScaleOut[row];
            Out[(size_t)row * HIDF + col] = val;
        }
    } else {
#pragma unroll
        for (int v = 0; v < 8; ++v) {
            const int row = row0 + v + 8 * half;
            if (row < nrows) {
                float val = fmaxf(fmaf(acc[v], rowScaleIn[row], bcol), 0.0f);
                if (SCALE_OUT) val *= rowScaleOut[row];
                Out[(size_t)row * HIDF + col] = val;
            }
        }
    }
}

// ---------------- final projection ----------------
// out[row] = h[row,:] . fc_w + fc_b - thres ; one wave per row.
__global__ void __launch_bounds__(256)
final_proj_kernel(const float* __restrict__ h, const float* __restrict__ fc_w,
                  const float* __restrict__ fc_b, const float* __restrict__ thres,
                  float* __restrict__ out, int nrows) {
    int row  = blockIdx.x * 8 + (threadIdx.x >> 5);
    int lane = threadIdx.x & 31;
    if (row >= nrows) return;
    const float4 v = *(const float4*)&h[(size_t)row * HIDF + lane * 4];
    const float4 w = *(const float4*)&fc_w[lane * 4];
    float p = v.x * w.x + v.y * w.y + v.z * w.z + v.w * w.w;
#pragma unroll
    for (int off = 16; off > 0; off >>= 1) p += __shfl_down(p, off, 32);
    if (lane == 0) out[row] = p + fc_b[0] - thres[0];
}

// ---------------- host-side orchestration ----------------

extern "C" void kernel_launch(void* const* d_in, const int* in_sizes, int n_in,
                              void* d_out, int out_size, void* d_ws, size_t ws_size,
                              hipStream_t stream) {
    const float* features = (const float*)d_in[0];
    const int*   src      = (const int*)d_in[1];
    const int*   dst      = (const int*)d_in[2];
    const float* W1       = (const float*)d_in[3];
    const float* b1       = (const float*)d_in[4];
    const float* W2       = (const float*)d_in[5];
    const float* b2       = (const float*)d_in[6];
    const float* fc_w     = (const float*)d_in[7];
    const float* fc_b     = (const float*)d_in[8];
    const float* thres    = (const float*)d_in[9];

    const int N = in_sizes[0] / HIDF;   // 50000
    const int E = in_sizes[1];          // 800000

    // workspace layout (floats): bufA[N*128] | bufB[N*128] | degO[N] | degI[N]
    float* ws   = (float*)d_ws;
    float* bufA = ws;                          // h1n -> h2
    float* bufB = bufA + (size_t)N * HIDF;     // agg1 -> agg2
    float* degO = bufB + (size_t)N * HIDF;     // deg_out -> dout_is
    float* degI = degO + N;                    // deg_in  -> din_is

    const int feat_n = N * HIDF;
    const dim3 blk(256);

    // 1) degree norms
    fill_zero_kernel<<<dim3((2 * N + 255) / 256), blk, 0, stream>>>(degO, 2 * N);
    degree_kernel<<<dim3((E + 255) / 256), blk, 0, stream>>>(src, dst, degO, degI, E);
    rsqrt_clip_kernel<<<dim3((2 * N + 255) / 256), blk, 0, stream>>>(degO, 2 * N);

    // 2) layer 1: agg1 = scatter(features * dout_is) ;
    //    h1n = relu(din_is * (agg1 @ W1) + b1) * dout_is
    fill_zero_kernel<<<dim3((feat_n + 255) / 256), blk, 0, stream>>>(bufB, feat_n);
    scatter_add_kernel<true><<<dim3((E + 7) / 8), blk, 0, stream>>>(
        src, dst, features, degO, bufB, E);
    gemm16_wmma_kernel<true><<<dim3((N + 15) / 16), blk, 0, stream>>>(
        bufB, degI, W1, b1, /*rowScaleOut=*/degO, bufA, N);

    // 3) layer 2: agg2 = scatter(h1n) ; h2 = relu(din_is * (agg2 @ W2) + b2)
    fill_zero_kernel<<<dim3((feat_n + 255) / 256), blk, 0, stream>>>(bufB, feat_n);
    scatter_add_kernel<false><<<dim3((E + 7) / 8), blk, 0, stream>>>(
        src, dst, bufA, nullptr, bufB, E);
    gemm16_wmma_kernel<false><<<dim3((N + 15) / 16), blk, 0, stream>>>(
        bufB, degI, W2, b2, /*rowScaleOut=*/nullptr, bufA, N);

    // 4) logits
    final_proj_kernel<<<dim3((N + 7) / 8), blk, 0, stream>>>(
        bufA, fc_w, fc_b, thres, (float*)d_out, N);
}